// LSTMAnswerDecoder_33973191312027
// MI455X (gfx1250) — compile-verified
//
#include <hip/hip_runtime.h>
#include <hip/hip_bf16.h>
#include <math.h>

typedef __attribute__((ext_vector_type(2))) float v2f;
typedef __attribute__((ext_vector_type(8))) float v8f;

// ---------------------------------------------------------------------------
// fp32 WMMA GEMM: D[M,N] = act( A[M,K]@B[K,N] (+ A2@B2) + bias (+bias2) )
// One wave (32 lanes) owns a 32x16 output strip: two 16x16 V_WMMA_F32_16X16X4_F32
// accumulator chains sharing one B fragment per K-step (2 loads / WMMA).
// Requires M % 32 == 0 (true for all GEMMs here), N % 16 == 0, K % 16 == 0.
// A-fragment (ISA 7.12.2, 32-bit A 16x4): lanes 0-15 K={0,1}, lanes 16-31 K={2,3}
// B-fragment: VGPR0 = rows K{0,2}, VGPR1 = rows K{1,3} striped over lanes.
// C/D: 8 VGPRs; lanes 0-15 -> M=vgpr, lanes 16-31 -> M=vgpr+8.
// ---------------------------------------------------------------------------
struct AccPair { v8f lo, hi; };

__device__ __forceinline__ AccPair wmma_acc2_f32(const float* __restrict__ A, int lda,
                                                 const float* __restrict__ B, int ldb,
                                                 int K, int m0, int n0,
                                                 int half, int r, AccPair acc) {
  const float* arow0 = A + (size_t)(m0 + r) * lda + 2 * half;        // M tile 0
  const float* arow1 = arow0 + (size_t)16 * lda;                     // M tile 1
  const float* bcol  = B + (size_t)(2 * half) * ldb + n0 + r;
#pragma unroll 4
  for (int k = 0; k < K; k += 4) {
    v2f a0 = *(const v2f*)(arow0 + k);
    v2f a1 = *(const v2f*)(arow1 + k);
    v2f b;
    b.x = bcol[(size_t)k * ldb];
    b.y = bcol[(size_t)(k + 1) * ldb];
    acc.lo = __builtin_amdgcn_wmma_f32_16x16x4_f32(
        false, a0, false, b, (short)0, acc.lo, false, false);
    acc.hi = __builtin_amdgcn_wmma_f32_16x16x4_f32(
        false, a1, false, b, (short)0, acc.hi, false, false);
  }
  return acc;
}

__global__ void wmma_gemm_f32(const float* __restrict__ A, int lda,
                              const float* __restrict__ Bm, int ldb, int K,
                              const float* __restrict__ A2, int lda2,
                              const float* __restrict__ B2, int ldb2, int K2,
                              const float* __restrict__ bias,
                              const float* __restrict__ bias2,
                              float* __restrict__ D, int ldd, int act) {
  const int n0 = blockIdx.x * 16;
  const int m0 = blockIdx.y * 32;
  const int lane = threadIdx.x;
  const int half = lane >> 4;   // 0: lanes 0-15, 1: lanes 16-31
  const int r = lane & 15;      // A-tile row / B-tile col within tile

  AccPair acc;
  acc.lo = (v8f){};
  acc.hi = (v8f){};
  acc = wmma_acc2_f32(A, lda, Bm, ldb, K, m0, n0, half, r, acc);
  if (A2) acc = wmma_acc2_f32(A2, lda2, B2, ldb2, K2, m0, n0, half, r, acc);

  float bv = 0.0f;
  if (bias)  bv += bias[n0 + r];
  if (bias2) bv += bias2[n0 + r];

  float* drow0 = D + (size_t)(m0 + 8 * half) * ldd + n0 + r;
  float* drow1 = drow0 + (size_t)16 * ldd;
#pragma unroll
  for (int i = 0; i < 8; ++i) {
    float v0 = acc.lo[i] + bv;
    float v1 = acc.hi[i] + bv;
    if (act) { v0 = tanhf(v0); v1 = tanhf(v1); }
    drow0[(size_t)i * ldd] = v0;
    drow1[(size_t)i * ldd] = v1;
  }
}

// ---------------------------------------------------------------------------
// ys[t][b][:] = embedding[t==0 ? START : answers[b][t-1]]     (T=64,B=32,E=256)
// ---------------------------------------------------------------------------
__global__ void build_ys(const float* __restrict__ emb, const int* __restrict__ ans,
                         float* __restrict__ ys) {
  int idx = blockIdx.x * 256 + threadIdx.x;      // total 64*32*256 = 524288
  int e = idx & 255;
  int b = (idx >> 8) & 31;
  int t = idx >> 13;
  int tok = (t == 0) ? 1 : ans[b * 64 + (t - 1)];   // START_IDX = 1
  ys[idx] = emb[(size_t)tok * 256 + e];
}

// ---------------------------------------------------------------------------
// ybar[b][0:256] = y_t[b], ybar[b][256:768] = o[b]
// ---------------------------------------------------------------------------
__global__ void pack_ybar(const float* __restrict__ yt, const float* __restrict__ o,
                          float* __restrict__ ybar) {
  int idx = blockIdx.x * 256 + threadIdx.x;      // total 32*768 = 24576
  int b = idx / 768, j = idx - b * 768;
  ybar[idx] = (j < 256) ? yt[b * 256 + j] : o[b * 512 + (j - 256)];
}

// ---------------------------------------------------------------------------
// Per-batch-row block: LSTM pointwise, soft attention over R=49 regions,
// writes u = [a | h_new], updates h/c in place.
// ---------------------------------------------------------------------------
__global__ void lstm_attn(const float* __restrict__ gates,
                          float* __restrict__ c, float* __restrict__ h,
                          const float* __restrict__ attn_proj,
                          const float* __restrict__ X,
                          float* __restrict__ u) {
  const int b = blockIdx.x;
  const int tid = threadIdx.x;
  __shared__ float s_h[512];
  __shared__ float s_e[64];

  // LSTM gates: i=[0:512) f=[512:1024) g=[1024:1536) o=[1536:2048)
  const float* grow = gates + (size_t)b * 2048;
  for (int j = tid; j < 512; j += 256) {
    float ig = grow[j];
    float fg = grow[512 + j];
    float gg = grow[1024 + j];
    float og = grow[1536 + j];
    float si = 1.0f / (1.0f + expf(-ig));
    float sf = 1.0f / (1.0f + expf(-fg));
    float so = 1.0f / (1.0f + expf(-og));
    float cn = sf * c[b * 512 + j] + si * tanhf(gg);
    float hn = so * tanhf(cn);
    c[b * 512 + j] = cn;
    h[b * 512 + j] = hn;
    s_h[j] = hn;
    u[(size_t)b * 1536 + 1024 + j] = hn;
  }
  __syncthreads();

  // e[r] = <attn_proj[b,r,:], h_new>, one wave per region (8 waves round-robin)
  const int wave = tid >> 5, lane = tid & 31;
  for (int r = wave; r < 49; r += 8) {
    const float* ap = attn_proj + ((size_t)b * 49 + r) * 512;
    float s = 0.0f;
    for (int k = lane; k < 512; k += 32) s += ap[k] * s_h[k];
    for (int off = 16; off > 0; off >>= 1) s += __shfl_xor(s, off, 32);
    if (lane == 0) s_e[r] = s;
  }
  __syncthreads();

  // softmax over 49 regions (tiny; thread 0 serial)
  if (tid == 0) {
    float m = s_e[0];
    for (int r = 1; r < 49; ++r) m = fmaxf(m, s_e[r]);
    float ssum = 0.0f;
    for (int r = 0; r < 49; ++r) { float v = expf(s_e[r] - m); s_e[r] = v; ssum += v; }
    float inv = 1.0f / ssum;
    for (int r = 0; r < 49; ++r) s_e[r] *= inv;
  }
  __syncthreads();

  // a[f] = sum_r att[r] * X[b,r,f]
  for (int f = tid; f < 1024; f += 256) {
    float s = 0.0f;
    for (int r = 0; r < 49; ++r) s += s_e[r] * X[((size_t)b * 49 + r) * 1024 + f];
    u[(size_t)b * 1536 + f] = s;
  }
}

// ---------------------------------------------------------------------------
static void gemm(hipStream_t s, const float* A, int lda, const float* B, int ldb, int K,
                 const float* A2, int lda2, const float* B2, int ldb2, int K2,
                 const float* bias, const float* bias2,
                 float* D, int ldd, int M, int N, int act) {
  dim3 grid(N / 16, M / 32);   // wave owns 32x16 strip
  wmma_gemm_f32<<<grid, 32, 0, s>>>(A, lda, B, ldb, K, A2, lda2, B2, ldb2, K2,
                                    bias, bias2, D, ldd, act);
}

extern "C" void kernel_launch(void* const* d_in, const int* in_sizes, int n_in,
                              void* d_out, int out_size, void* d_ws, size_t ws_size,
                              hipStream_t stream) {
  const float* X      = (const float*)d_in[0];   // [32,49,1024]
  const float* Gfeat  = (const float*)d_in[1];   // [32,2048]
  const float* Q      = (const float*)d_in[2];   // [32,512]
  const int*   ans    = (const int*)  d_in[3];   // [32,64]
  const float* emb    = (const float*)d_in[4];   // [10000,256]
  const float* Wg2o_w = (const float*)d_in[5];   // [2048,512]
  const float* Wg2o_b = (const float*)d_in[6];   // [512]
  const float* Wh     = (const float*)d_in[7];   // [512,512]
  const float* Wc     = (const float*)d_in[8];   // [512,512]
  const float* Wih    = (const float*)d_in[9];   // [768,2048]
  const float* Whh    = (const float*)d_in[10];  // [512,2048]
  const float* bih    = (const float*)d_in[11];  // [2048]
  const float* bhh    = (const float*)d_in[12];  // [2048]
  const float* Wattn  = (const float*)d_in[13];  // [1024,512]
  const float* Wu_w   = (const float*)d_in[14];  // [1536,512]
  const float* Wu_b   = (const float*)d_in[15];  // [512]
  const float* Wv_w   = (const float*)d_in[16];  // [512,10000]
  const float* Wv_b   = (const float*)d_in[17];  // [10000]
  float* out = (float*)d_out;                    // [32,64,10000]

  // workspace carve-out (floats)
  float* ws    = (float*)d_ws;
  float* o     = ws;                 // 32*512
  float* h     = o     + 32 * 512;   // 32*512
  float* c     = h     + 32 * 512;   // 32*512
  float* ybar  = c     + 32 * 512;   // 32*768
  float* gates = ybar  + 32 * 768;   // 32*2048
  float* u     = gates + 32 * 2048;  // 32*1536
  float* ys    = u     + 32 * 1536;  // 64*32*256
  float* ap    = ys    + 64 * 32 * 256; // 32*49*512

  // ---- step-invariant precompute ----
  // o0 = Gfeat @ Wg2o_w + b
  gemm(stream, Gfeat, 2048, Wg2o_w, 512, 2048, nullptr, 0, nullptr, 0, 0,
       Wg2o_b, nullptr, o, 512, 32, 512, 0);
  // h0 = Q @ Wh ; c0 = Q @ Wc
  gemm(stream, Q, 512, Wh, 512, 512, nullptr, 0, nullptr, 0, 0,
       nullptr, nullptr, h, 512, 32, 512, 0);
  gemm(stream, Q, 512, Wc, 512, 512, nullptr, 0, nullptr, 0, 0,
       nullptr, nullptr, c, 512, 32, 512, 0);
  // attn_proj = X(view [1568,1024]) @ Wattn   -> [1568,512]
  gemm(stream, X, 1024, Wattn, 512, 1024, nullptr, 0, nullptr, 0, 0,
       nullptr, nullptr, ap, 512, 1568, 512, 0);
  // teacher-forced inputs
  build_ys<<<2048, 256, 0, stream>>>(emb, ans, ys);

  // ---- T=64 recurrent steps ----
  for (int t = 0; t < 64; ++t) {
    pack_ybar<<<96, 256, 0, stream>>>(ys + (size_t)t * 32 * 256, o, ybar);
    // gates = ybar@Wih + h@Whh + bih + bhh   [32,2048]
    gemm(stream, ybar, 768, Wih, 2048, 768, h, 512, Whh, 2048, 512,
         bih, bhh, gates, 2048, 32, 2048, 0);
    lstm_attn<<<32, 256, 0, stream>>>(gates, c, h, ap, X, u);
    // o_new = tanh(u @ Wu_w + Wu_b)  [32,512]  (o no longer needed after gates)
    gemm(stream, u, 1536, Wu_w, 512, 1536, nullptr, 0, nullptr, 0, 0,
         Wu_b, nullptr, o, 512, 32, 512, 1);
    // logits_t = o_new @ Wv_w + Wv_b -> out[:, t, :]  (row stride T*V)
    gemm(stream, o, 512, Wv_w, 10000, 512, nullptr, 0, nullptr, 0, 0,
         Wv_b, nullptr, out + (size_t)t * 10000, 64 * 10000, 32, 10000, 0);
  }
}